// SpectralProjection_42657615184051
// MI455X (gfx1250) — compile-verified
//
#include <hip/hip_runtime.h>
#include <math.h>

#define B_        16
#define T_        2048
#define NFREQ_    1152
#define NFREQ_OCT 128
#define EPS_      1e-4f
#define TWO_PI_   6.28318530717958647692f

typedef float v2f __attribute__((ext_vector_type(2)));
typedef float v8f __attribute__((ext_vector_type(8)));

// ---------------------------------------------------------------------------
// Stage 1: one block = (batch b, one octave = 128 freqs). 256 threads = 8
// waves; each wave owns 16 frequencies and runs the full T=2048 reduction
// through V_WMMA_F32_16X16X4_F32 accumulators (fp32-exact).
// ---------------------------------------------------------------------------
__global__ __launch_bounds__(256)
void spectral_mag_kernel(const float* __restrict__ batch,
                         const float* __restrict__ freqs,
                         float* __restrict__ magbuf) {
    __shared__ float s_ts[T_];
    __shared__ float s_yn[T_];
    __shared__ float s_red[32];

    const int b    = blockIdx.y;
    const int oct  = blockIdx.x;
    const int tid  = threadIdx.x;
    const int lane = tid & 31;
    const int wave = tid >> 5;

    const float* ts = batch + (size_t)b * 2 * T_;
    const float* ys = ts + T_;

    // ---- stage ts/ys into LDS, accumulate sum & sumsq of ys ----
    float sum = 0.f, sumsq = 0.f;
    for (int i = tid; i < T_; i += 256) {
        float tv = ts[i];
        float yv = ys[i];
        s_ts[i] = tv;
        s_yn[i] = yv;               // raw; normalized below
        sum   += yv;
        sumsq += yv * yv;
    }
    // wave32 shuffle reduction
    for (int off = 16; off > 0; off >>= 1) {
        sum   += __shfl_down(sum,   off, 32);
        sumsq += __shfl_down(sumsq, off, 32);
    }
    if (lane == 0) { s_red[wave] = sum; s_red[8 + wave] = sumsq; }
    __syncthreads();
    if (tid == 0) {
        float s = 0.f, s2 = 0.f;
        for (int w = 0; w < 8; ++w) { s += s_red[w]; s2 += s_red[8 + w]; }
        float mean = s / (float)T_;
        float var  = (s2 - s * s / (float)T_) / (float)(T_ - 1);  // ddof=1
        float sd   = sqrtf(fmaxf(var, 0.f));
        s_red[16] = mean;
        s_red[17] = 1.f / (sd + EPS_);
    }
    __syncthreads();
    const float mean = s_red[16];
    const float rstd = s_red[17];
    for (int i = tid; i < T_; i += 256)
        s_yn[i] = (s_yn[i] - mean) * rstd;
    __syncthreads();

    // ---- main loop: WMMA-accumulated projection ----
    // A (16x4 f32) layout: lanes 0-15 -> M=lane, K={0,1}; lanes 16-31 -> M=lane-16, K={2,3}
    // B (4x16 f32) layout mirrors K striping; we replicate yn[t] across all columns.
    const int   m     = lane & 15;
    const int   kbase = (lane >> 4) << 1;            // 0 or 2
    const int   fbase = oct * NFREQ_OCT + wave * 16;
    const float omega = TWO_PI_ * freqs[fbase + m];

    v8f acc_s = {};   // p1 accumulator (every column identical)
    v8f acc_c = {};   // p2 accumulator

    for (int t0 = 0; t0 < T_; t0 += 4) {
        const float tv0 = s_ts[t0 + kbase];
        const float tv1 = s_ts[t0 + kbase + 1];
        const float y0  = s_yn[t0 + kbase];
        const float y1  = s_yn[t0 + kbase + 1];

        float s0, c0, s1, c1;
        sincosf(tv0 * omega, &s0, &c0);
        sincosf(tv1 * omega, &s1, &c1);

        v2f a_sin = {s0, s1};
        v2f a_cos = {c0, c1};
        v2f bv    = {y0, y1};

        acc_s = __builtin_amdgcn_wmma_f32_16x16x4_f32(
            false, a_sin, false, bv, (short)0, acc_s, false, false);
        acc_c = __builtin_amdgcn_wmma_f32_16x16x4_f32(
            false, a_cos, false, bv, (short)0, acc_c, false, false);
    }

    // ---- extract: C/D layout: lanes 0-15 col N=lane, VGPR v -> M=v;
    //              lanes 16-31 col N=lane-16, VGPR v -> M=v+8.
    // All columns equal; lanes 0-7 emit M=0..7, lanes 16-23 emit M=8..15.
    union { v8f v; float f[8]; } us, uc;
    us.v = acc_s;
    uc.v = acc_c;

    const bool active = (lane < 8) || (lane >= 16 && lane < 24);
    if (active) {
        const int mi = (lane < 8) ? lane : (lane - 16 + 8);
        const float p1 = us.f[lane & 7];
        const float p2 = uc.f[lane & 7];
        const float mag = sqrtf(p1 * p1 + p2 * p2);
        magbuf[b * NFREQ_ + fbase + mi] = mag;
    }
}

// ---------------------------------------------------------------------------
// Stage 2: per-batch tnorm over the 1152 magnitudes. One block per batch.
// ---------------------------------------------------------------------------
__global__ __launch_bounds__(256)
void spectral_tnorm_kernel(const float* __restrict__ magbuf,
                           float* __restrict__ out) {
    __shared__ float s_red[32];
    const int b    = blockIdx.x;
    const int tid  = threadIdx.x;
    const int lane = tid & 31;
    const int wave = tid >> 5;
    const float* mrow = magbuf + b * NFREQ_;

    float sum = 0.f, sumsq = 0.f;
    for (int i = tid; i < NFREQ_; i += 256) {
        float v = mrow[i];
        sum   += v;
        sumsq += v * v;
    }
    for (int off = 16; off > 0; off >>= 1) {
        sum   += __shfl_down(sum,   off, 32);
        sumsq += __shfl_down(sumsq, off, 32);
    }
    if (lane == 0) { s_red[wave] = sum; s_red[8 + wave] = sumsq; }
    __syncthreads();
    if (tid == 0) {
        float s = 0.f, s2 = 0.f;
        for (int w = 0; w < 8; ++w) { s += s_red[w]; s2 += s_red[8 + w]; }
        float meanv = s / (float)NFREQ_;
        float var   = (s2 - s * s / (float)NFREQ_) / (float)(NFREQ_ - 1);  // ddof=1
        float sd    = sqrtf(fmaxf(var, 0.f));
        s_red[16] = meanv;
        s_red[17] = 1.f / (sd + EPS_);
    }
    __syncthreads();
    const float meanv = s_red[16];
    const float rstd  = s_red[17];
    for (int i = tid; i < NFREQ_; i += 256)
        out[b * NFREQ_ + i] = (mrow[i] - meanv) * rstd;
}

extern "C" void kernel_launch(void* const* d_in, const int* in_sizes, int n_in,
                              void* d_out, int out_size, void* d_ws, size_t ws_size,
                              hipStream_t stream) {
    const float* batch = (const float*)d_in[0];   // (16, 2, 2048) f32
    const float* freqs = (const float*)d_in[1];   // (1152,) f32
    float* out    = (float*)d_out;                // (16, 1, 1152) f32
    float* magbuf = (float*)d_ws;                 // 16*1152 f32 scratch

    dim3 grid1(NFREQ_ / NFREQ_OCT, B_);           // 9 octaves x 16 batches
    spectral_mag_kernel<<<grid1, 256, 0, stream>>>(batch, freqs, magbuf);
    spectral_tnorm_kernel<<<B_, 256, 0, stream>>>(magbuf, out);
}